// SimulationDecoder_5669356833838
// MI455X (gfx1250) — compile-verified
//
#include <hip/hip_runtime.h>

typedef __attribute__((ext_vector_type(2))) float v2f;
typedef __attribute__((ext_vector_type(8))) float v8f;

#define NB    128        // batches
#define NP    32         // particles (== wave32 lanes)
#define NT    50         // time samples
#define NE    (NP*(NP-1))
#define NSTEP 100        // SAMPLE_FREQ
#define DTs   0.001f
#define MAXF  100.0f     // 0.1 / DT
#define EPSk  1e-10f
#define BOXk  5.0f

// v_readlane_b32 broadcast (uniform literal lane) -> SGPR scalar operand
__device__ __forceinline__ float rlane(float v, int l) {
    return __builtin_bit_cast(float,
        __builtin_amdgcn_readlane(__builtin_bit_cast(int, v), l));
}

// lane i <-> lane i^16 swap via v_permlanex16_b32 (identity selects), pure VALU
__device__ __forceinline__ float swap16(float v) {
    int iv = __builtin_bit_cast(int, v);
    int r  = __builtin_amdgcn_permlanex16(iv, iv, 0x76543210, 0xFEDCBA98,
                                          false, false);
    return __builtin_bit_cast(float, r);
}

// One wave simulates one frame (b,t). Lane = particle index.
// Gram matrix loc*loc^T done with 4x V_WMMA_F32_16X16X4_F32 per step.
__global__ __launch_bounds__(256) void SimulationDecoder_kernel(
    const float4* __restrict__ inp,   // (B, N, T, 4)  loc.xy, vel.xy
    const float*  __restrict__ rel,   // (B, E, 2)
    float4*       __restrict__ out)   // (B, N, T-1, 4)
{
    const int lane = threadIdx.x & 31;
    const int wave = blockIdx.x * (blockDim.x >> 5) + (threadIdx.x >> 5);
    const int F = NB * (NT - 1);
    if (wave >= F) return;                       // wave-uniform; EXEC stays full
    const int b = wave / (NT - 1);
    const int t = wave % (NT - 1);

    // ---- load state: loc = 5*in, vel = 5*in (the 0.5*(x+1)*10-5 collapse) ----
    float4 s = inp[(b * NP + lane) * NT + t];
    float x  = 5.f * s.x, y  = 5.f * s.y;
    float vx = 5.f * s.z, vy = 5.f * s.w;

    // ---- fsz[i][j] = 2*rel[b, e(i,j), 1] - 1, row-major off-diagonal edges ----
    float fsz[NP];
    #pragma unroll
    for (int j = 0; j < NP; ++j) {
        if (j == lane) {
            fsz[j] = -1.f;                        // diagonal: contributes 0 (dx=dy=0)
        } else {
            int e = lane * (NP - 1) + j - (j > lane ? 1 : 0);
            fsz[j] = 2.f * rel[(b * NE + e) * 2 + 1] - 1.f;
        }
    }

    const bool lo = (lane < 16);

    #pragma unroll 1
    for (int step = 0; step < NSTEP; ++step) {
        // ---- Gram matrix G = loc * loc^T via 4x v_wmma_f32_16x16x4_f32 ----
        // A 16x4 layout: lanes0-15 hold K0,K1 (rows m=lane); lanes16-31 hold K2,K3 (=0).
        // B 4x16 layout mirrors it, so the symmetric operand uses identical registers.
        float xh = swap16(x);
        float yh = swap16(y);
        v2f P, Q;
        P.x = lo ? x  : 0.f;  P.y = lo ? y  : 0.f;   // rows/cols 0..15
        Q.x = lo ? xh : 0.f;  Q.y = lo ? yh : 0.f;   // rows/cols 16..31
        v8f z = {0.f, 0.f, 0.f, 0.f, 0.f, 0.f, 0.f, 0.f};

        float g[NP];   // g[j] = G[j][lane] = loc_j . loc_lane  (G symmetric)
        {   // M-tile 0 (rows 0..15): D(0,0), D(0,1)
            v8f D0 = __builtin_amdgcn_wmma_f32_16x16x4_f32(false, P, false, P, (short)0, z, false, false);
            v8f D1 = __builtin_amdgcn_wmma_f32_16x16x4_f32(false, P, false, Q, (short)0, z, false, false);
            #pragma unroll
            for (int rr = 0; rr < 8; ++rr) {
                float u = D0[rr], v = D1[rr];
                float us = swap16(u);
                float vs = swap16(v);
                g[rr]     = lo ? u  : vs;   // row rr
                g[rr + 8] = lo ? us : v;    // row rr+8
            }
        }
        {   // M-tile 1 (rows 16..31): D(1,0), D(1,1)
            v8f D0 = __builtin_amdgcn_wmma_f32_16x16x4_f32(false, Q, false, P, (short)0, z, false, false);
            v8f D1 = __builtin_amdgcn_wmma_f32_16x16x4_f32(false, Q, false, Q, (short)0, z, false, false);
            #pragma unroll
            for (int rr = 0; rr < 8; ++rr) {
                float u = D0[rr], v = D1[rr];
                float us = swap16(u);
                float vs = swap16(v);
                g[rr + 16] = lo ? u  : vs;
                g[rr + 24] = lo ? us : v;
            }
        }

        // ---- all-pairs forces: sq = |r_i + r_j - 2 G_ij| (matches reference) ----
        float r  = fmaf(x, x, y * y);
        float fx = 0.f, fy = 0.f;
        #pragma unroll
        for (int j = 0; j < NP; ++j) {
            float xj = rlane(x, j);                // v_readlane broadcasts
            float yj = rlane(y, j);
            float rj = rlane(r, j);                // reuse per-lane r, as reference does
            float dx = x - xj, dy = y - yj;
            float sq = fabsf(fmaf(-2.f, g[j], rj) + r);
            // sq^1.5 via raw 1-ulp hardware transcendentals (v_sqrt_f32 / v_rcp_f32);
            // the +EPS guard keeps the operands well-conditioned.
            float l2p3 = sq * __builtin_amdgcn_sqrtf(sq);
            float fs = fsz[j] * __builtin_amdgcn_rcpf(l2p3 + EPSk);
            fx = fmaf(fs, dx, fx);
            fy = fmaf(fs, dy, fy);
        }
        fx = fminf(fmaxf(fx, -MAXF), MAXF);
        fy = fminf(fmaxf(fy, -MAXF), MAXF);

        // ---- integrate + reflect at box walls (over first, then under) ----
        vx = fmaf(DTs, fx, vx);  vy = fmaf(DTs, fy, vy);
        x  = fmaf(DTs, vx, x);   y  = fmaf(DTs, vy, y);
        {
            bool ov = x >  BOXk; x = ov ?  2.f*BOXk - x : x; vx = ov ? -fabsf(vx) : vx;
            bool un = x < -BOXk; x = un ? -2.f*BOXk - x : x; vx = un ?  fabsf(vx) : vx;
        }
        {
            bool ov = y >  BOXk; y = ov ?  2.f*BOXk - y : y; vy = ov ? -fabsf(vy) : vy;
            bool un = y < -BOXk; y = un ? -2.f*BOXk - y : y; vy = un ?  fabsf(vy) : vy;
        }
    }

    // ---- renormalize and store (B, N, T-1, 4) ----
    float4 o;
    o.x = x  * 0.2f;  o.y = y  * 0.2f;
    o.z = vx * 0.2f;  o.w = vy * 0.2f;
    out[(b * NP + lane) * (NT - 1) + t] = o;
}

extern "C" void kernel_launch(void* const* d_in, const int* in_sizes, int n_in,
                              void* d_out, int out_size, void* d_ws, size_t ws_size,
                              hipStream_t stream) {
    const float4* inp = (const float4*)d_in[0];   // inputs (128,32,50,4) f32
    const float*  rel = (const float*)d_in[1];    // relations (128,992,2) f32
    // d_in[2], d_in[3] (rel_rec, rel_send) unused by the reference
    float4* out = (float4*)d_out;                 // (128,32,49,4) f32

    const int F = NB * (NT - 1);                  // 6272 frames, 1 wave each
    const int wavesPerBlock = 256 / 32;           // 8 waves / block
    const int blocks = (F + wavesPerBlock - 1) / wavesPerBlock;  // 784
    SimulationDecoder_kernel<<<blocks, 256, 0, stream>>>(inp, rel, out);
}